// RotationLearner_52450140619045
// MI455X (gfx1250) — compile-verified
//
#include <hip/hip_runtime.h>

// ---------------------------------------------------------------------------
// expm(skew(params)) for N=1024 via scaling-and-squaring + Taylor-Horner,
// all GEMMs on CDNA5 fp32 WMMA (v_wmma_f32_16x16x4_f32), double-buffered LDS.
// ---------------------------------------------------------------------------

typedef float v2f __attribute__((ext_vector_type(2)));
typedef float v8f __attribute__((ext_vector_type(8)));

constexpr int NDIM = 1024;
constexpr int BT   = 128;            // block tile (rows and cols)
constexpr int KT   = 32;             // K chunk staged in LDS
constexpr int ASTR = KT + 4;         // 36 floats/row: 16B-aligned, conflict-free A frags
constexpr int BSTR = BT + 8;         // 136 floats/row: half-waves hit disjoint bank windows
constexpr int ABUF = BT * ASTR;      // floats per A LDS buffer
constexpr int BBUF = KT * BSTR;      // floats per B LDS buffer
constexpr int HBUF = ABUF + BBUF;    // one double-buffer half
constexpr int LDS_FLOATS = 2 * HBUF;

// ---- build B = skew(params) * 2^-s ----------------------------------------
__global__ __launch_bounds__(256)
void skew_build_kernel(const float* __restrict__ p, float* __restrict__ A, float s)
{
    const int j = blockIdx.x * blockDim.x + threadIdx.x;  // column
    const int i = blockIdx.y;                             // row
    float v;
    if (i == j) {
        v = 0.0f;
    } else if (i < j) {
        const int k = i * (NDIM - 1) - (i * (i - 1)) / 2 + (j - i - 1);
        v = p[k] * s;
    } else {
        const int k = j * (NDIM - 1) - (j * (j - 1)) / 2 + (i - j - 1);
        v = -p[k] * s;
    }
    A[i * NDIM + j] = v;
}

// ---- T = I + c*B (first Horner step, elementwise) -------------------------
__global__ __launch_bounds__(256)
void taylor_init_kernel(const float* __restrict__ B, float* __restrict__ T, float c)
{
    const int j = blockIdx.x * blockDim.x + threadIdx.x;
    const int i = blockIdx.y;
    T[i * NDIM + j] = B[i * NDIM + j] * c + ((i == j) ? 1.0f : 0.0f);
}

// ---- staging helpers -------------------------------------------------------
__device__ __forceinline__ void stage_load(const float* __restrict__ A,
                                           const float* __restrict__ Bm,
                                           int blockM, int blockN, int k0, int tid,
                                           float4 (&aReg)[4], float4 (&bReg)[4])
{
#pragma unroll
    for (int r = 0; r < 4; ++r) {                // A tile: 128 rows x 32 k
        const int idx = tid + r * 256;
        const int m   = idx >> 3;
        const int kc  = idx & 7;
        aReg[r] = *reinterpret_cast<const float4*>(&A[(blockM + m) * NDIM + k0 + kc * 4]);
    }
#pragma unroll
    for (int r = 0; r < 4; ++r) {                // B tile: 32 k x 128 cols
        const int idx = tid + r * 256;
        const int k   = idx >> 5;
        const int nc  = idx & 31;
        bReg[r] = *reinterpret_cast<const float4*>(&Bm[(k0 + k) * NDIM + blockN + nc * 4]);
    }
}

__device__ __forceinline__ void stage_store(float* __restrict__ As, float* __restrict__ Bs,
                                            int tid,
                                            const float4 (&aReg)[4], const float4 (&bReg)[4])
{
#pragma unroll
    for (int r = 0; r < 4; ++r) {
        const int idx = tid + r * 256;
        const int m   = idx >> 3;
        const int kc  = idx & 7;
        *reinterpret_cast<float4*>(&As[m * ASTR + kc * 4]) = aReg[r];
    }
#pragma unroll
    for (int r = 0; r < 4; ++r) {
        const int idx = tid + r * 256;
        const int k   = idx >> 5;
        const int nc  = idx & 31;
        *reinterpret_cast<float4*>(&Bs[k * BSTR + nc * 4]) = bReg[r];
    }
}

__device__ __forceinline__ void compute_chunk(const float* __restrict__ As,
                                              const float* __restrict__ Bs,
                                              int wm, int wn, int h, int l16,
                                              v8f (&acc)[2][4])
{
#pragma unroll
    for (int kk = 0; kk < KT; kk += 4) {
        // A fragment: lane = M; VGPR v holds K = kk + 2h + v  (one b64 per tile row)
        v2f af[2];
#pragma unroll
        for (int i = 0; i < 2; ++i)
            af[i] = *reinterpret_cast<const v2f*>(
                &As[(wm + i * 16 + l16) * ASTR + kk + 2 * h]);

        // B fragment: lane = N; VGPR v holds K-row kk + 2h + v
        v2f bf[4];
#pragma unroll
        for (int j = 0; j < 4; ++j) {
            const int n = wn + j * 16 + l16;
            bf[j].x = Bs[(kk + 2 * h + 0) * BSTR + n];
            bf[j].y = Bs[(kk + 2 * h + 1) * BSTR + n];
        }

#pragma unroll
        for (int i = 0; i < 2; ++i)
#pragma unroll
            for (int j = 0; j < 4; ++j)
                acc[i][j] = __builtin_amdgcn_wmma_f32_16x16x4_f32(
                    false, af[i], false, bf[j], (short)0, acc[i][j], false, false);
    }
}

// ---- D = scale*(A @ B) [+ I]  ---------------------------------------------
// 128x128 block tile, 8 waves (4 rows x 2 cols), each wave: 2x4 tiles of 16x16.
// Double-buffered LDS: chunk c+1 global loads issue before chunk c's 64 WMMAs.
__global__ __launch_bounds__(256)
void gemm_wmma_f32_kernel(const float* __restrict__ A, const float* __restrict__ Bm,
                          float* __restrict__ D, float scale, int addIdentity)
{
    extern __shared__ float smem[];

    const int tid  = threadIdx.x;
    const int lane = tid & 31;
    const int wave = tid >> 5;
    const int h    = lane >> 4;      // half-wave select
    const int l16  = lane & 15;

    const int wm = (wave >> 1) * 32; // wave row origin inside block tile
    const int wn = (wave & 1) * 64;  // wave col origin inside block tile

    const int blockM = blockIdx.y * BT;
    const int blockN = blockIdx.x * BT;

    v8f acc[2][4];
    const v8f vzero = {0.f, 0.f, 0.f, 0.f, 0.f, 0.f, 0.f, 0.f};
#pragma unroll
    for (int i = 0; i < 2; ++i)
#pragma unroll
        for (int j = 0; j < 4; ++j)
            acc[i][j] = vzero;

    float4 aReg[4], bReg[4];
    stage_load(A, Bm, blockM, blockN, 0, tid, aReg, bReg);
    stage_store(smem, smem + ABUF, tid, aReg, bReg);
    __syncthreads();

    constexpr int NC = NDIM / KT;    // 32 chunks
    for (int c = 0; c < NC; ++c) {
        const int cur = c & 1;
        const int nxt = cur ^ 1;
        // LDS double-buffer halves selected by pointer arithmetic (no pointer
        // arrays: addrspace(3) pointer tables break the static initializer path).
        float* curBase = smem + cur * HBUF;
        float* nxtBase = smem + nxt * HBUF;

        if (c + 1 < NC)  // issue next chunk's global loads before the WMMA block
            stage_load(A, Bm, blockM, blockN, (c + 1) * KT, tid, aReg, bReg);

        compute_chunk(curBase, curBase + ABUF, wm, wn, h, l16, acc);

        if (c + 1 < NC)  // safe: buffer `nxt` was last read in chunk c-1 (barriered)
            stage_store(nxtBase, nxtBase + ABUF, tid, aReg, bReg);
        __syncthreads();
    }

    // Epilogue: C/D layout -> lane = N, VGPR v: M = v + 8h. Fuse scale and +I.
#pragma unroll
    for (int i = 0; i < 2; ++i) {
#pragma unroll
        for (int j = 0; j < 4; ++j) {
            const int col  = blockN + wn + j * 16 + l16;
            const int row0 = blockM + wm + i * 16 + 8 * h;
#pragma unroll
            for (int v = 0; v < 8; ++v) {
                const int row = row0 + v;
                float val = acc[i][j][v] * scale;
                if (addIdentity && (row == col)) val += 1.0f;
                D[row * NDIM + col] = val;
            }
        }
    }
}

extern "C" void kernel_launch(void* const* d_in, const int* in_sizes, int n_in,
                              void* d_out, int out_size, void* d_ws, size_t ws_size,
                              hipStream_t stream) {
    (void)in_sizes; (void)n_in; (void)out_size; (void)ws_size;

    const float* params = (const float*)d_in[0];
    float* out = (float*)d_out;

    const size_t N2 = (size_t)NDIM * NDIM;
    float* B  = (float*)d_ws;        // scaled skew matrix, 4 MB
    float* T0 = B + N2;              // ping-pong temp,     4 MB
    // d_out doubles as the second ping-pong buffer.

    constexpr int SQUARINGS = 10;    // scale by 2^-10: ||B||_2 ~ 0.06
    constexpr int TAYLOR_K  = 16;    // Horner degree; remainder << f32 eps

    const dim3 eltBlock(256), eltGrid(NDIM / 256, NDIM);
    const dim3 gemmBlock(256), gemmGrid(NDIM / BT, NDIM / BT);
    const size_t shmemBytes = (size_t)LDS_FLOATS * sizeof(float);  // 71,680 B

    // B = skew(params) * 2^-SQUARINGS
    skew_build_kernel<<<eltGrid, eltBlock, 0, stream>>>(params, B, 1.0f / 1024.0f);

    // T = I + B/K
    taylor_init_kernel<<<eltGrid, eltBlock, 0, stream>>>(B, T0, 1.0f / (float)TAYLOR_K);

    // Horner: T <- I + (1/k) * B @ T,  k = K-1 .. 1   (15 GEMMs)
    float* src = T0;
    float* dst = out;
    for (int k = TAYLOR_K - 1; k >= 1; --k) {
        gemm_wmma_f32_kernel<<<gemmGrid, gemmBlock, shmemBytes, stream>>>(
            B, src, dst, 1.0f / (float)k, 1);
        float* t = src; src = dst; dst = t;
    }

    // Squaring: E <- E @ E, 10 times. 15 + 10 = 25 writes (odd) -> ends in d_out.
    for (int s = 0; s < SQUARINGS; ++s) {
        gemm_wmma_f32_kernel<<<gemmGrid, gemmBlock, shmemBytes, stream>>>(
            src, src, dst, 1.0f, 0);
        float* t = src; src = dst; dst = t;
    }
    // Final result now resides in d_out (src == out).
}